// RetrievalerLoss_49993419325680
// MI455X (gfx1250) — compile-verified
//
#include <hip/hip_runtime.h>

// ---------------------------------------------------------------------------
// Problem constants (match reference)
// ---------------------------------------------------------------------------
#define BB     4096      // batch
#define DD     256       // embedding dim (= K for every GEMM here)
#define DHH    256       // head dim
#define CC     32        // candidates per chunk
#define HH     2         // hard chunks
#define RR     1         // random chunks
#define NCHUNK (1 + HH + RR)
#define NROWS  (BB * NCHUNK)     // 16384
#define NSEL   (HH * CC)         // need top-64 hard negatives
#define MARGIN 0.5f
#define NEG_INF (-3.402823466e38f)

typedef float v2f __attribute__((ext_vector_type(2)));
typedef float v8f __attribute__((ext_vector_type(8)));

// ---------------------------------------------------------------------------
// Deterministic hash PRNG (surrogate for jax threefry — compile-only env)
// ---------------------------------------------------------------------------
__device__ __forceinline__ unsigned int hash_u32(unsigned int a, unsigned int b,
                                                 unsigned int c) {
  unsigned int h = a * 0x9E3779B1u ^ (b + 0x85EBCA77u) * 0xC2B2AE3Du ^ c * 0x27D4EB2Fu;
  h ^= h >> 16; h *= 0x7FEB352Du;
  h ^= h >> 15; h *= 0x846CA68Bu;
  h ^= h >> 16;
  return h;
}
__device__ __forceinline__ float hash_f01(unsigned int a, unsigned int b,
                                          unsigned int c) {
  return (float)(hash_u32(a, b, c) >> 8) * (1.0f / 16777216.0f);
}

// ---------------------------------------------------------------------------
// K1: row-normalize embeddings (for sim = nrm @ nrm^T)
// ---------------------------------------------------------------------------
__global__ __launch_bounds__(256) void normalize_kernel(
    const float* __restrict__ emb, float* __restrict__ nrm) {
  const int i = blockIdx.x, t = threadIdx.x;
  __shared__ float red[256];
  float v = emb[(size_t)i * DD + t];
  red[t] = v * v;
  __syncthreads();
  for (int off = 128; off > 0; off >>= 1) {
    if (t < off) red[t] += red[t + off];
    __syncthreads();
  }
  float inv = 1.0f / sqrtf(red[0]);
  nrm[(size_t)i * DD + t] = v * inv;
}

// ---------------------------------------------------------------------------
// K2: f32 WMMA GEMM with async-LDS A staging.
//   C[M,N] = A[M,256] @ B[256,N]      (transB: B is [N,256] row-major)
// Block = 256 threads = 8 waves. One block owns a 16-row A strip (mt) and 8
// adjacent 16-col tiles (nt). The 16x256 f32 A strip (16 KB) is staged into
// LDS once per block with GLOBAL_LOAD_ASYNC_TO_LDS_B128 (ASYNCcnt +
// s_wait_asynccnt, per ISA ch.10.7/15.18), then each wave runs the K-loop with
// V_WMMA_F32_16X16X4_F32: A fragments come from LDS (ds_load), B fragments
// from global (L2-resident). Per-lane layouts follow ISA 7.12.2:
//   A 16x4 : lane l -> row l%16, K = (l/16)*2 .. +1          (v2f)
//   B 4x16 : lane l -> col l%16, K = (l/16)*2 .. +1          (v2f)
//   C 16x16: lane l -> col l%16, VGPR r -> row r + 8*(l/16)  (v8f)
// EXEC is all-ones: exact grids, no divergent branches.
// ---------------------------------------------------------------------------
__global__ __launch_bounds__(256) void gemm_f32_wmma_kernel(
    const float* __restrict__ A, const float* __restrict__ B,
    float* __restrict__ Cm, int N, int transB) {
  __shared__ float sA[16 * DD];  // 16 KB: rows mt*16..+15, K = 0..255
  const int t = threadIdx.x;
  const int wave = t >> 5;
  const int lane = t & 31;
  const int l16  = lane & 15;
  const int half = lane >> 4;
  const int ngrp = N >> 7;                    // (N/16)/8 tile-groups per row
  const int mt = blockIdx.x / ngrp;
  const int nt = (blockIdx.x - mt * ngrp) * 8 + wave;

  // ---- async stage of the A strip: 1024 x 16B chunks, 4 per thread ----
  {
    const unsigned lds_base = (unsigned)(uintptr_t)(&sA[0]);
    const char* gbase = (const char*)(A + (size_t)(mt << 4) * DD);
#pragma unroll
    for (int c2 = 0; c2 < 4; ++c2) {
      const int ch  = t + 256 * c2;           // chunk id 0..1023
      const int row = ch >> 6;                // 64 chunks (1 KB) per row
      const int off = (ch & 63) << 4;         // byte offset within row
      const unsigned lds_addr = lds_base + (unsigned)(ch << 4);
      const unsigned long long gaddr =
          (unsigned long long)(uintptr_t)(gbase + (size_t)row * (DD * 4) + off);
      asm volatile("global_load_async_to_lds_b128 %0, %1, off"
                   :: "v"(lds_addr), "v"(gaddr) : "memory");
    }
    asm volatile("s_wait_asynccnt 0" ::: "memory");
  }
  __syncthreads();

  v8f acc = {0.f, 0.f, 0.f, 0.f, 0.f, 0.f, 0.f, 0.f};
  const float* aptr = &sA[l16 * DD + half * 2];

  if (transB) {  // B given as [N,256] row-major: column n of B^T is row n of B
    const float* brow = B + (size_t)((nt << 4) + l16) * DD + half * 2;
    for (int kk = 0; kk < DD; kk += 4) {
      const float2 af = *(const float2*)(aptr + kk);   // ds_load_b64
      const float2 bf = *(const float2*)(brow + kk);   // global_load_b64
      v2f a, b;
      a[0] = af.x; a[1] = af.y;
      b[0] = bf.x; b[1] = bf.y;
      acc = __builtin_amdgcn_wmma_f32_16x16x4_f32(false, a, false, b,
                                                  (short)0, acc, false, false);
    }
  } else {
    const int col = (nt << 4) + l16;
    for (int kk = 0; kk < DD; kk += 4) {
      const float2 af = *(const float2*)(aptr + kk);
      v2f a, b;
      a[0] = af.x; a[1] = af.y;
      b[0] = B[(size_t)(kk + half * 2) * N + col];
      b[1] = B[(size_t)(kk + half * 2 + 1) * N + col];
      acc = __builtin_amdgcn_wmma_f32_16x16x4_f32(false, a, false, b,
                                                  (short)0, acc, false, false);
    }
  }
#pragma unroll
  for (int r = 0; r < 8; ++r) {
    Cm[(size_t)((mt << 4) + r + (half << 3)) * N + (nt << 4) + l16] = acc[r];
  }
}

// ---------------------------------------------------------------------------
// K3: dust = emb @ w_db + b_db   (one row per wave)
// ---------------------------------------------------------------------------
__global__ __launch_bounds__(256) void dust_kernel(
    const float* __restrict__ emb, const float* __restrict__ w_db,
    const float* __restrict__ b_db, float* __restrict__ dust) {
  const int wave = threadIdx.x >> 5, lane = threadIdx.x & 31;
  const int row = blockIdx.x * 8 + wave;
  float s = 0.f;
  for (int k = lane; k < DD; k += 32) s += emb[(size_t)row * DD + k] * w_db[k];
  for (int off = 16; off > 0; off >>= 1) s += __shfl_xor(s, off, 32);
  if (lane == 0) dust[row] = s + b_db[0];
}

// ---------------------------------------------------------------------------
// K4: per-row chunk construction.
// One block per query row i: neg-count, pos choice, top-64 hard negatives
// (stable argmax: greater value, tie -> smaller index, matching stable
// argsort), top-32 random negatives (hash-keyed), then fill the 4 chunk rows
// (pos, hard x2, random) with Fisher-Yates shuffles of the valid prefix.
// ---------------------------------------------------------------------------
__global__ __launch_bounds__(256) void build_rows_kernel(
    const float* __restrict__ sim, const unsigned char* __restrict__ posm,
    const unsigned char* __restrict__ negm, int* __restrict__ c_idx,
    unsigned char* __restrict__ c_maskb, int* __restrict__ targets,
    int* __restrict__ ppos, unsigned char* __restrict__ is_pos) {
  __shared__ float vals[BB];
  __shared__ unsigned char nmask[BB];
  __shared__ float red_v[256];
  __shared__ int red_i[256];
  __shared__ int s_hard[NSEL];
  __shared__ int s_rnd[CC];
  __shared__ int s_negcount;
  __shared__ int s_poschoice;

  const int i = blockIdx.x, t = threadIdx.x;

  // ---- prologue: masks, masked sim row, neg count, positive choice ----
  int localneg = 0;
  float bestpv = -1.0f;
  int bestpi = 0x7fffffff;
  for (int j = t; j < BB; j += 256) {
    const unsigned char ng = (j != i) ? negm[(size_t)i * BB + j] : (unsigned char)0;
    const unsigned char ps = (j != i) ? posm[(size_t)i * BB + j] : (unsigned char)0;
    nmask[j] = ng;
    vals[j] = ng ? sim[(size_t)i * BB + j] : NEG_INF;
    localneg += ng ? 1 : 0;
    if (ps) {
      const float r = hash_f01(0xA11CE5u, (unsigned)i, (unsigned)j);
      if (r > bestpv || (r == bestpv && j < bestpi)) { bestpv = r; bestpi = j; }
    }
  }
  red_i[t] = localneg;
  __syncthreads();
  for (int off = 128; off > 0; off >>= 1) {
    if (t < off) red_i[t] += red_i[t + off];
    __syncthreads();
  }
  if (t == 0) s_negcount = red_i[0];
  __syncthreads();
  red_v[t] = bestpv;
  red_i[t] = bestpi;
  __syncthreads();
  for (int off = 128; off > 0; off >>= 1) {
    if (t < off) {
      const float v2 = red_v[t + off];
      const int i2 = red_i[t + off];
      if (v2 > red_v[t] || (v2 == red_v[t] && i2 < red_i[t])) {
        red_v[t] = v2; red_i[t] = i2;
      }
    }
    __syncthreads();
  }
  if (t == 0) s_poschoice = (red_v[0] < 0.f) ? ((i + 1) & (BB - 1)) : red_i[0];
  __syncthreads();

  // ---- top-64 hardest negatives by sim (extract-max loop) ----
  for (int s = 0; s < NSEL; ++s) {
    float bv = NEG_INF;
    int bi = 0x7fffffff;
    for (int j = t; j < BB; j += 256) {
      const float v = vals[j];
      if (v > bv || (v == bv && j < bi)) { bv = v; bi = j; }
    }
    red_v[t] = bv; red_i[t] = bi;
    __syncthreads();
    for (int off = 128; off > 0; off >>= 1) {
      if (t < off) {
        const float v2 = red_v[t + off];
        const int i2 = red_i[t + off];
        if (v2 > red_v[t] || (v2 == red_v[t] && i2 < red_i[t])) {
          red_v[t] = v2; red_i[t] = i2;
        }
      }
      __syncthreads();
    }
    if (t == 0) {
      const int w = (red_i[0] < BB) ? red_i[0] : 0;
      s_hard[s] = w;
      vals[w] = NEG_INF;
    }
    __syncthreads();
  }

  // ---- top-32 random negatives (hash-keyed, ~= random permutation prefix) --
  for (int j = t; j < BB; j += 256)
    vals[j] = nmask[j] ? hash_f01(0xD1CEu, (unsigned)i, (unsigned)j) : NEG_INF;
  __syncthreads();
  for (int s = 0; s < CC; ++s) {
    float bv = NEG_INF;
    int bi = 0x7fffffff;
    for (int j = t; j < BB; j += 256) {
      const float v = vals[j];
      if (v > bv || (v == bv && j < bi)) { bv = v; bi = j; }
    }
    red_v[t] = bv; red_i[t] = bi;
    __syncthreads();
    for (int off = 128; off > 0; off >>= 1) {
      if (t < off) {
        const float v2 = red_v[t + off];
        const int i2 = red_i[t + off];
        if (v2 > red_v[t] || (v2 == red_v[t] && i2 < red_i[t])) {
          red_v[t] = v2; red_i[t] = i2;
        }
      }
      __syncthreads();
    }
    if (t == 0) {
      const int w = (red_i[0] < BB) ? red_i[0] : 0;
      s_rnd[s] = w;
      vals[w] = NEG_INF;
    }
    __syncthreads();
  }

  // ---- serial epilogue: assemble + shuffle the 4 chunk rows ----
  if (t == 0) {
    const int negc = s_negcount;
    const int pc = s_poschoice;
    int tmp[CC];

    // chunk 0: positive + hardest C-1
    tmp[0] = pc;
    for (int c = 1; c < CC; ++c) tmp[c] = s_hard[c - 1];
    int valid = 1 + (negc < CC - 1 ? negc : CC - 1);
    for (int c = valid; c < CC; ++c) tmp[c] = 0;
    for (int k = valid - 1; k > 0; --k) {
      const unsigned int r = hash_u32(0x51u, (unsigned)i, (unsigned)k) % (unsigned)(k + 1);
      const int sw = tmp[k]; tmp[k] = tmp[r]; tmp[r] = sw;
    }
    int ppv = 0;
    for (int c = 0; c < valid; ++c)
      if (tmp[c] == pc) { ppv = c; break; }
    {
      const size_t row = (size_t)i;
      for (int c = 0; c < CC; ++c) {
        c_idx[row * CC + c] = tmp[c];
        c_maskb[row * CC + c] = (unsigned char)(c < valid);
      }
      targets[row] = ppv + 1;
      ppos[row] = ppv;
      is_pos[row] = 1;
    }

    // hard chunks
    for (int n = 0; n < HH; ++n) {
      const int start = n * CC;
      int vc = negc - start;
      vc = vc < 0 ? 0 : (vc > CC ? CC : vc);
      int base = start;
      if (vc == 0) { base = 0; vc = negc < CC ? negc : CC; }
      for (int c = 0; c < CC; ++c) tmp[c] = (c < vc) ? s_hard[base + c] : 0;
      for (int k = vc - 1; k > 0; --k) {
        const unsigned int r =
            hash_u32(0x52u + (unsigned)n, (unsigned)i, (unsigned)k) % (unsigned)(k + 1);
        const int sw = tmp[k]; tmp[k] = tmp[r]; tmp[r] = sw;
      }
      const size_t row = (size_t)(1 + n) * BB + i;
      for (int c = 0; c < CC; ++c) {
        c_idx[row * CC + c] = tmp[c];
        c_maskb[row * CC + c] = (unsigned char)(c < vc);
      }
      targets[row] = 0; ppos[row] = -1; is_pos[row] = 0;
    }

    // random chunk
    {
      const int vc = negc < CC ? negc : CC;
      for (int c = 0; c < CC; ++c) tmp[c] = (c < vc) ? s_rnd[c] : 0;
      for (int k = vc - 1; k > 0; --k) {
        const unsigned int r = hash_u32(0x53u, (unsigned)i, (unsigned)k) % (unsigned)(k + 1);
        const int sw = tmp[k]; tmp[k] = tmp[r]; tmp[r] = sw;
      }
      const size_t row = (size_t)(1 + HH) * BB + i;
      for (int c = 0; c < CC; ++c) {
        c_idx[row * CC + c] = tmp[c];
        c_maskb[row * CC + c] = (unsigned char)(c < vc);
      }
      targets[row] = 0; ppos[row] = -1; is_pos[row] = 0;
    }
  }
}

// ---------------------------------------------------------------------------
// K5: cand_logits[n,c] = (qh[n%B] . chall[c_idx[n,c]]) / 16 (masked -> -1e9)
// ---------------------------------------------------------------------------
__global__ __launch_bounds__(256) void cand_logits_kernel(
    const float* __restrict__ qh, const float* __restrict__ chall,
    const int* __restrict__ c_idx, const unsigned char* __restrict__ c_maskb,
    float* __restrict__ cand_logits) {
  const int wave = threadIdx.x >> 5, lane = threadIdx.x & 31;
  const int n = blockIdx.x;
  const float* qr = qh + (size_t)(n & (BB - 1)) * DHH;
  for (int c = wave; c < CC; c += 8) {
    const int idx = c_idx[(size_t)n * CC + c];
    const float* cr = chall + (size_t)idx * DHH;
    __builtin_prefetch(cr, 0, 1);  // global_prefetch_b8
    float s = 0.f;
    for (int k = lane; k < DHH; k += 32) s += qr[k] * cr[k];
    for (int off = 16; off > 0; off >>= 1) s += __shfl_xor(s, off, 32);
    if (lane == 0) {
      float lg = s * 0.0625f;  // 1/sqrt(256)
      if (!c_maskb[(size_t)n * CC + c]) lg = -1.0e9f;
      cand_logits[(size_t)n * CC + c] = lg;
    }
  }
}

// ---------------------------------------------------------------------------
// K6: per-row loss pieces -> atomic accumulators
// accum: 0=ce_sum 1=db_sum 2=npos 3=rank_sum 4=vr_cnt 5=db2_sum 6=ve_cnt
// ---------------------------------------------------------------------------
__global__ __launch_bounds__(256) void loss_kernel(
    const float* __restrict__ cand_logits, const unsigned char* __restrict__ c_maskb,
    const float* __restrict__ dust, const int* __restrict__ targets,
    const int* __restrict__ ppos, const unsigned char* __restrict__ is_pos,
    float* __restrict__ accum) {
  const int wave = threadIdx.x >> 5, lane = threadIdx.x & 31;
  const int n = blockIdx.x * 8 + wave;

  const float lg = cand_logits[(size_t)n * CC + lane];  // lane == candidate c
  const int mk = c_maskb[(size_t)n * CC + lane];
  const float du = dust[n & (BB - 1)];
  const int tgt = targets[n];
  const int ip = is_pos[n];
  const int pp = ip ? ppos[n] : 0;

  // logsumexp over [dust, cand_logits]
  float m = lg;
  for (int off = 16; off > 0; off >>= 1) m = fmaxf(m, __shfl_xor(m, off, 32));
  m = fmaxf(m, du);
  float e = expf(lg - m);
  for (int off = 16; off > 0; off >>= 1) e += __shfl_xor(e, off, 32);

  const float pv = __shfl(lg, pp, 32);                      // logit at pos slot
  const float tc = __shfl(lg, (tgt > 0) ? (tgt - 1) : 0, 32);
  float mn = (mk && !(ip && lane == pp)) ? lg : NEG_INF;    // max over nm
  for (int off = 16; off > 0; off >>= 1) mn = fmaxf(mn, __shfl_xor(mn, off, 32));
  float me = mk ? (lg - du) : NEG_INF;                      // max delta over mask
  for (int off = 16; off > 0; off >>= 1) me = fmaxf(me, __shfl_xor(me, off, 32));

  if (lane == 0) {
    const float logZ = m + logf(e + expf(du - m));
    const float tlg = (tgt == 0) ? du : tc;
    atomicAdd(&accum[0], logZ - tlg);  // ce numerator
    if (ip) {
      atomicAdd(&accum[1], fmaxf(0.f, MARGIN - (pv - du)));  // db pos term
      atomicAdd(&accum[2], 1.0f);                            // npos
      if (mn > -1.0e30f) {
        atomicAdd(&accum[3], fmaxf(0.f, MARGIN + mn - pv));  // rank term
        atomicAdd(&accum[4], 1.0f);                          // vr count
      }
    } else {
      if (me > -1.0e30f) {
        atomicAdd(&accum[5], fmaxf(0.f, MARGIN + me));       // db neg term
        atomicAdd(&accum[6], 1.0f);                          // ve count
      }
    }
  }
}

__global__ void zero_accum_kernel(float* __restrict__ accum) {
  if (threadIdx.x < 8) accum[threadIdx.x] = 0.f;
}

__global__ void finalize_kernel(const float* __restrict__ accum,
                                float* __restrict__ out) {
  if (threadIdx.x == 0) {
    const float ce = accum[0] / (float)NROWS;
    const float db = accum[1] / fmaxf(accum[2], 1.f) + accum[5] / fmaxf(accum[6], 1.f);
    const float rk = accum[3] / fmaxf(accum[4], 1.f);
    out[0] = ce + 1.0f * db + 1.0f * rk;  // L_DB = L_RANK = 1.0
  }
}

// ---------------------------------------------------------------------------
// Host launcher
// ---------------------------------------------------------------------------
extern "C" void kernel_launch(void* const* d_in, const int* in_sizes, int n_in,
                              void* d_out, int out_size, void* d_ws, size_t ws_size,
                              hipStream_t stream) {
  const float* emb = (const float*)d_in[0];
  const float* Wq  = (const float*)d_in[1];
  const float* Wc  = (const float*)d_in[2];
  const float* wdb = (const float*)d_in[3];
  const float* bdb = (const float*)d_in[4];
  const unsigned char* posm = (const unsigned char*)d_in[5];
  const unsigned char* negm = (const unsigned char*)d_in[6];

  char* ws = (char*)d_ws;
  size_t o = 0;
  auto carve = [&](size_t bytes) -> char* {
    char* p = ws + o;
    o = (o + bytes + 255) & ~(size_t)255;
    return p;
  };
  float* sim   = (float*)carve((size_t)BB * BB * sizeof(float));   // 64 MB
  float* nrm   = (float*)carve((size_t)BB * DD * sizeof(float));
  float* qh    = (float*)carve((size_t)BB * DHH * sizeof(float));
  float* chall = (float*)carve((size_t)BB * DHH * sizeof(float));
  float* dust  = (float*)carve((size_t)BB * sizeof(float));
  int*   c_idx = (int*)carve((size_t)NROWS * CC * sizeof(int));
  unsigned char* c_maskb = (unsigned char*)carve((size_t)NROWS * CC);
  int*   targets = (int*)carve((size_t)NROWS * sizeof(int));
  int*   ppv     = (int*)carve((size_t)NROWS * sizeof(int));
  unsigned char* is_pos = (unsigned char*)carve((size_t)NROWS);
  float* cand_logits = (float*)carve((size_t)NROWS * CC * sizeof(float));
  float* accum = (float*)carve(32 * sizeof(float));
  if (o > ws_size) return;  // workspace too small — nothing safe to do

  zero_accum_kernel<<<1, 32, 0, stream>>>(accum);
  normalize_kernel<<<BB, 256, 0, stream>>>(emb, nrm);

  // sim = nrm @ nrm^T : 256 mt-strips x 32 nt-groups (8 tiles each)
  gemm_f32_wmma_kernel<<<(BB / 16) * (BB / 128), 256, 0, stream>>>(
      nrm, nrm, sim, BB, 1);
  // qh = emb @ Wq, chall = emb @ Wc : 256 mt-strips x 2 nt-groups
  gemm_f32_wmma_kernel<<<(BB / 16) * (DHH / 128), 256, 0, stream>>>(
      emb, Wq, qh, DHH, 0);
  gemm_f32_wmma_kernel<<<(BB / 16) * (DHH / 128), 256, 0, stream>>>(
      emb, Wc, chall, DHH, 0);

  dust_kernel<<<BB / 8, 256, 0, stream>>>(emb, wdb, bdb, dust);
  build_rows_kernel<<<BB, 256, 0, stream>>>(sim, posm, negm, c_idx, c_maskb,
                                            targets, ppv, is_pos);
  cand_logits_kernel<<<NROWS, 256, 0, stream>>>(qh, chall, c_idx, c_maskb,
                                                cand_logits);
  loss_kernel<<<NROWS / 8, 256, 0, stream>>>(cand_logits, c_maskb, dust,
                                             targets, ppv, is_pos, accum);
  finalize_kernel<<<1, 32, 0, stream>>>(accum, (float*)d_out);
}